// Voxelization_79456894976548
// MI455X (gfx1250) — compile-verified
//
#include <hip/hip_runtime.h>
#include <hip/hip_bf16.h>
#include <stdint.h>

// Problem constants (match reference setup_inputs / init_kwargs)
#define RES    32
#define R3     (RES * RES * RES)   // 32768
#define BATCH  8
#define CHN    64
#define NPTS   100000
#define TILE   256
#define NTILES ((NPTS + TILE - 1) / TILE)   // 391
#define OUTSZ  ((size_t)BATCH * CHN * R3)   // 16,777,216

// ---------------- CDNA5 feature guards ----------------
#if defined(__AMDGCN__) && defined(__has_builtin)
#if __has_builtin(__builtin_amdgcn_global_load_async_to_lds_b32)
#define HAVE_ASYNC_LDS 1
#endif
#endif
#ifndef HAVE_ASYNC_LDS
#define HAVE_ASYNC_LDS 0
#endif

#if defined(__AMDGCN__)
typedef __attribute__((address_space(1))) int as1_int;
typedef __attribute__((address_space(3))) int as3_int;
#endif

__device__ __forceinline__ void wait_async0() {
#if defined(__AMDGCN__)
#if defined(__has_builtin) && __has_builtin(__builtin_amdgcn_s_wait_asynccnt)
  __builtin_amdgcn_s_wait_asynccnt(0);
#else
  asm volatile("s_wait_asynccnt 0" ::: "memory");
#endif
#endif
}

// Native fp32 global atomic add, no return (lowers to global_atomic_add_f32)
__device__ __forceinline__ void atomAddF32(float* p, float v) {
  __hip_atomic_fetch_add(p, v, __ATOMIC_RELAXED, __HIP_MEMORY_SCOPE_AGENT);
}

// ---------------- Kernel 0: zero output + counts ----------------
__global__ __launch_bounds__(256) void vox_init(float* __restrict__ out,
                                                float* __restrict__ cnt) {
  size_t t = (size_t)blockIdx.x * blockDim.x + threadIdx.x;
  if (t < OUTSZ) out[t] = 0.0f;
  if (t < (size_t)BATCH * R3) cnt[t] = 0.0f;
}

// ---------------- Kernel 1: voxel index + valid count ----------------
__global__ __launch_bounds__(256) void vox_index(const float* __restrict__ coords,
                                                 int* __restrict__ idxArr,
                                                 float* __restrict__ cnt) {
  int t = blockIdx.x * blockDim.x + threadIdx.x;
  if (t >= BATCH * NPTS) return;
  int b = t / NPTS;
  const float* p = coords + (size_t)t * 3;
  float cx = (p[0] + 1.0f) * 0.5f;
  float cy = (p[1] + 1.0f) * 0.5f;
  float cz = (p[2] + 1.0f) * 0.5f;
  bool valid = (cx >= 0.0f) & (cx <= 1.0f) &
               (cy >= 0.0f) & (cy <= 1.0f) &
               (cz >= 0.0f) & (cz <= 1.0f);
  // jnp.round == round-half-to-even == rintf under default RTNE
  int i = (int)rintf(cx * (float)(RES - 1));
  int j = (int)rintf(cy * (float)(RES - 1));
  int k = (int)rintf(cz * (float)(RES - 1));
  int idx = i * (RES * RES) + j * RES + k;
  idx = min(max(idx, 0), R3 - 1);
  idxArr[t] = valid ? idx : -1;
  if (valid) atomAddF32(cnt + (size_t)b * R3 + idx, 1.0f);
}

// ---------------- Kernel 2: scatter-add features ----------------
// One block = one (batch, 256-point tile). Index tile staged to LDS once via
// the CDNA5 async global->LDS path, then reused for all 64 channels.
__global__ __launch_bounds__(256) void vox_scatter(const float* __restrict__ feats,
                                                   const int* __restrict__ idxArr,
                                                   float* __restrict__ out) {
  __shared__ int sIdx[TILE];
  const int b    = blockIdx.x / NTILES;
  const int tile = blockIdx.x % NTILES;
  const int t    = threadIdx.x;
  const int n    = tile * TILE + t;
  const int nclamp = (n < NPTS) ? n : (NPTS - 1);
  const int* gp = idxArr + (size_t)b * NPTS + nclamp;

#if HAVE_ASYNC_LDS
  __builtin_amdgcn_global_load_async_to_lds_b32(
      (as1_int*)gp, (as3_int*)&sIdx[t], /*offset=*/0, /*cpol=*/0);
  wait_async0();
#else
  sIdx[t] = *gp;
#endif
  __syncthreads();

  const int e = (n < NPTS) ? sIdx[t] : -1;
  const size_t fbase = ((size_t)b * CHN) * NPTS + n;  // + c*NPTS (coalesced in n)
  const size_t obase = ((size_t)b * CHN) * R3;        // + c*R3 + e
  if (e >= 0) {
#pragma unroll 4
    for (int c = 0; c < CHN; ++c) {
      float f = feats[fbase + (size_t)c * NPTS];
      atomAddF32(out + obase + (size_t)c * R3 + e, f);
    }
  }
}

// ---------------- Kernel 3: normalize by count ----------------
__global__ __launch_bounds__(256) void vox_norm(float* __restrict__ out,
                                                const float* __restrict__ cnt) {
  size_t t = (size_t)blockIdx.x * blockDim.x + threadIdx.x;
  if (t >= OUTSZ) return;
  size_t v  = t & (size_t)(R3 - 1);
  size_t bc = t >> 15;          // /R3
  size_t b  = bc >> 6;          // /CHN
  float c = cnt[b * R3 + v];
  out[t] = out[t] / fmaxf(c, 1.0f);
}

extern "C" void kernel_launch(void* const* d_in, const int* in_sizes, int n_in,
                              void* d_out, int out_size, void* d_ws, size_t ws_size,
                              hipStream_t stream) {
  const float* feats  = (const float*)d_in[0];   // [B,C,N] f32
  const float* coords = (const float*)d_in[1];   // [B,N,3] f32
  float* out = (float*)d_out;                    // [B,C,R,R,R] f32

  // Workspace: counts [B*R3] f32, then indices [B*N] i32  (~4.3 MB total)
  float* cnt   = (float*)d_ws;
  int*   idxAr = (int*)((char*)d_ws + (size_t)BATCH * R3 * sizeof(float));

  {
    size_t total = OUTSZ;
    int blocks = (int)((total + 255) / 256);
    vox_init<<<blocks, 256, 0, stream>>>(out, cnt);
  }
  {
    int total = BATCH * NPTS;
    vox_index<<<(total + 255) / 256, 256, 0, stream>>>(coords, idxAr, cnt);
  }
  vox_scatter<<<BATCH * NTILES, 256, 0, stream>>>(feats, idxAr, out);
  {
    int blocks = (int)((OUTSZ + 255) / 256);
    vox_norm<<<blocks, 256, 0, stream>>>(out, cnt);
  }
}